// GNN_25589415149714
// MI455X (gfx1250) — compile-verified
//
#include <hip/hip_runtime.h>
#include <hip/hip_bf16.h>
#include <math.h>

#define N_NODES 100000
#define N_EDGES 1600000
#define HID     64
#define OUT_DIM 40

typedef __attribute__((ext_vector_type(2))) float v2f;
typedef __attribute__((ext_vector_type(8))) float v8f;

// Native float atomic add (global_atomic_add_f32, no CAS loop)
__device__ __forceinline__ void atomAddF(float* p, float v) {
    unsafeAtomicAdd(p, v);
}

// ---------------- degree / norm ----------------

__global__ void k_deg_init(float* deg, int n) {
    int i = blockIdx.x * blockDim.x + threadIdx.x;
    if (i < n) deg[i] = 1.0f;            // self-loop contributes 1 to every node
}

__global__ void k_deg_edges(const int* __restrict__ dst, float* deg, int e) {
    int i = blockIdx.x * blockDim.x + threadIdx.x;
    if (i < e) atomAddF(&deg[dst[i]], 1.0f);
}

__global__ void k_dinv(const float* __restrict__ deg, float* dinv, int n) {
    int i = blockIdx.x * blockDim.x + threadIdx.x;
    if (i < n) {
        float d = deg[i];
        dinv[i] = (d > 0.0f) ? rsqrtf(d) : 0.0f;
    }
}

// ---------------- fp32 WMMA GEMM: C[M x OUTN] = A[M x 64] @ B[64 x OUTN] (+bias) ----------------
// M must be a multiple of 16 (100000 = 6250*16). lda = 64, ldb = ldc = OUTN.
// B is staged into LDS transposed and zero-padded to 64x64 so the WMMA inner
// loop is branch-free: A fragment = global_load_b64, B fragment = ds_load_b64.

template <int OUTN>
__global__ void k_gemm_wmma_f32(const float* __restrict__ A,
                                const float* __restrict__ B,
                                const float* __restrict__ bias,
                                float* __restrict__ C, int M) {
    constexpr int NCT = (OUTN + 15) / 16;     // column tiles
    __shared__ float Bt[64 * 64];             // Bt[c*64 + k], cols >= OUTN are 0

    for (int lin = threadIdx.x; lin < 64 * 64; lin += blockDim.x) {
        int c = lin >> 6;
        int k = lin & 63;
        Bt[lin] = (c < OUTN) ? B[k * OUTN + c] : 0.0f;
    }
    __syncthreads();

    const int lane = threadIdx.x & 31;
    const int half = lane >> 4;               // 0: lanes 0-15, 1: lanes 16-31
    const int lm   = lane & 15;

    const int waveId = (blockIdx.x * blockDim.x + threadIdx.x) >> 5;
    const int nWaves = (gridDim.x * blockDim.x) >> 5;
    const int numRowTiles = M >> 4;
    const int totTiles    = numRowTiles * NCT;

    for (int tile = waveId; tile < totTiles; tile += nWaves) {
        const int rt   = tile / NCT;
        const int ct   = tile - rt * NCT;
        const int row0 = rt << 4;
        const int col0 = ct << 4;

        v8f c = {0.f, 0.f, 0.f, 0.f, 0.f, 0.f, 0.f, 0.f};

        // A fragment: lane (half,lm) holds A[row0+lm][k0 + 2*half + {0,1}]
        // B fragment: lane (half,lm) holds B[k0 + 2*half + {0,1}][col0+lm]
        const float* aptr = A  + (size_t)(row0 + lm) * 64 + 2 * half;
        const float* bptr = Bt + (size_t)(col0 + lm) * 64 + 2 * half;

#pragma unroll
        for (int k0 = 0; k0 < 64; k0 += 4) {
            v2f a = *(const v2f*)(aptr + k0);
            v2f b = *(const v2f*)(bptr + k0);
            c = __builtin_amdgcn_wmma_f32_16x16x4_f32(false, a, false, b,
                                                      (short)0, c, false, false);
        }

        // C/D layout: VGPR r -> row (row0 + r + 8*half), col (col0 + lm)
        const int bcol = col0 + lm;
        const bool bok = (OUTN % 16 == 0) ? true : (bcol < OUTN);
        if (bok) {
            float badd = bias ? bias[bcol] : 0.0f;
#pragma unroll
            for (int r = 0; r < 8; ++r) {
                int row = row0 + r + 8 * half;
                C[(size_t)row * OUTN + bcol] = c[r] + badd;
            }
        }
    }
}

// ---------------- graph aggregation ----------------

// h[i][:] = xw[i][:] * dinv[i]^2   (self-loop term; also initializes h)
__global__ void k_self(const float* __restrict__ xw, const float* __restrict__ dinv,
                       float* __restrict__ h, int n) {
    long long i = (long long)blockIdx.x * blockDim.x + threadIdx.x;
    if (i < (long long)n * HID) {
        int node = (int)(i >> 6);
        float w = dinv[node];
        h[i] = xw[i] * (w * w);
    }
}

// h[dst][:] += xw[src][:] * dinv[src]*dinv[dst], one thread per (edge, dim)
__global__ void k_edge_agg(const int* __restrict__ src, const int* __restrict__ dst,
                           const float* __restrict__ dinv,
                           const float* __restrict__ xw, float* __restrict__ h, int e) {
    long long t = (long long)blockIdx.x * blockDim.x + threadIdx.x;
    if (t < (long long)e * HID) {
        int ed = (int)(t >> 6);
        int j  = (int)(t & 63);
        int s  = src[ed];
        int d  = dst[ed];
        float w = dinv[s] * dinv[d];
        atomAddF(&h[(size_t)d * HID + j], xw[(size_t)s * HID + j] * w);
    }
}

__global__ void k_bias_relu(float* __restrict__ h, const float* __restrict__ bias, int n) {
    long long i = (long long)blockIdx.x * blockDim.x + threadIdx.x;
    if (i < (long long)n * HID) {
        float v = h[i] + bias[(int)(i & 63)];
        h[i] = fmaxf(v, 0.0f);
    }
}

// ---------------- log_softmax over 40 classes ----------------

__global__ void k_log_softmax(const float* __restrict__ logits, float* __restrict__ out, int n) {
    int i = blockIdx.x * blockDim.x + threadIdx.x;
    if (i < n) {
        const float* p = logits + (size_t)i * OUT_DIM;
        float m = -INFINITY;
#pragma unroll
        for (int j = 0; j < OUT_DIM; ++j) m = fmaxf(m, p[j]);
        float s = 0.0f;
#pragma unroll
        for (int j = 0; j < OUT_DIM; ++j) s += expf(p[j] - m);
        float ls = m + logf(s);
        float* o = out + (size_t)i * OUT_DIM;
#pragma unroll
        for (int j = 0; j < OUT_DIM; ++j) o[j] = p[j] - ls;
    }
}

// ---------------- driver ----------------

extern "C" void kernel_launch(void* const* d_in, const int* in_sizes, int n_in,
                              void* d_out, int out_size, void* d_ws, size_t ws_size,
                              hipStream_t stream) {
    const float* x  = (const float*)d_in[0];
    const int*   ei = (const int*)d_in[1];            // [2, E] row-major (int32)
    const float* W1 = (const float*)d_in[2];
    const float* b1 = (const float*)d_in[3];
    const float* W2 = (const float*)d_in[4];
    const float* b2 = (const float*)d_in[5];
    const float* W3 = (const float*)d_in[6];
    const float* b3 = (const float*)d_in[7];
    const float* W4 = (const float*)d_in[8];
    const float* b4 = (const float*)d_in[9];

    const int* src = ei;              // edge_index[0]
    const int* dst = ei + N_EDGES;    // edge_index[1]

    // workspace layout (floats): deg[N] | dinv[N] | bufA[N*64] | bufB[N*64]
    float* ws   = (float*)d_ws;
    float* deg  = ws;
    float* dinv = ws + N_NODES;
    float* bufA = ws + 2 * (size_t)N_NODES;
    float* bufB = bufA + (size_t)N_NODES * HID;
    float* out  = (float*)d_out;

    const int T = 256;
    const int gN   = (N_NODES + T - 1) / T;
    const int gE   = (N_EDGES + T - 1) / T;
    const int gNH  = (int)(((long long)N_NODES * HID + T - 1) / T);
    const int gEH  = (int)(((long long)N_EDGES * HID + T - 1) / T);
    const int gemmBlocks = 1024;      // 8192 waves, grid-strided over tiles

    // GCN normalization
    k_deg_init <<<gN, T, 0, stream>>>(deg, N_NODES);
    k_deg_edges<<<gE, T, 0, stream>>>(dst, deg, N_EDGES);
    k_dinv     <<<gN, T, 0, stream>>>(deg, dinv, N_NODES);

    // layer 1: h1 = relu(agg(x @ W1) + b1)
    k_gemm_wmma_f32<HID><<<gemmBlocks, T, 0, stream>>>(x, W1, nullptr, bufA, N_NODES);
    k_self     <<<gNH, T, 0, stream>>>(bufA, dinv, bufB, N_NODES);
    k_edge_agg <<<gEH, T, 0, stream>>>(src, dst, dinv, bufA, bufB, N_EDGES);
    k_bias_relu<<<gNH, T, 0, stream>>>(bufB, b1, N_NODES);

    // layer 2: h2 = relu(agg(h1 @ W2) + b2)
    k_gemm_wmma_f32<HID><<<gemmBlocks, T, 0, stream>>>(bufB, W2, nullptr, bufA, N_NODES);
    k_self     <<<gNH, T, 0, stream>>>(bufA, dinv, bufB, N_NODES);
    k_edge_agg <<<gEH, T, 0, stream>>>(src, dst, dinv, bufA, bufB, N_EDGES);
    k_bias_relu<<<gNH, T, 0, stream>>>(bufB, b2, N_NODES);

    // dense head: h3 = h2 @ W3 + b3 ; logits = h3 @ W4 + b4
    k_gemm_wmma_f32<HID>    <<<gemmBlocks, T, 0, stream>>>(bufB, W3, b3, bufA, N_NODES);
    k_gemm_wmma_f32<OUT_DIM><<<gemmBlocks, T, 0, stream>>>(bufA, W4, b4, bufB, N_NODES);

    k_log_softmax<<<gN, T, 0, stream>>>(bufB, out, N_NODES);
}